// DRL4Metro_30897994728262
// MI455X (gfx1250) — compile-verified
//
#include <hip/hip_runtime.h>
#include <hip/hip_bf16.h>
#include <math.h>

// Problem constants from the reference (station_num_lim is a device scalar we
// cannot read during graph capture; the reference fixes STEPS=20).
#define BSZ   8
#define SSZ   8192
#define HSZ   128
#define NSTEP 20

typedef __attribute__((ext_vector_type(2))) float v2f;
typedef __attribute__((ext_vector_type(8))) float v8f;

__device__ __forceinline__ float sigm_(float x) { return 1.0f / (1.0f + expf(-x)); }

// Hardware tanh (CDNA5 V_TANH_F32, a TRANS op). If the builtin is declared the
// compiler handles the TRANS-result hazard; otherwise inline asm with a v_nop
// filling the required 1-independent-op slot after the TRANS (ISA §7.4.3).
#if __has_builtin(__builtin_amdgcn_tanhf)
__device__ __forceinline__ float tanh_(float x) { return __builtin_amdgcn_tanhf(x); }
#else
__device__ __forceinline__ float tanh_(float x) {
  float r;
  asm volatile("v_tanh_f32 %0, %1\n\tv_nop" : "=v"(r) : "v"(x));
  return r;
}
#endif

// ---------------------------------------------------------------------------
// k_pre: fold W_ref@(W_s·)+W_pd@(W_d·) into rank-3 coefficients
//   A[g]  = sum_h W_ref[g,h] * W_s[h,0]
//   Bv[g] = sum_h W_ref[g,h] * W_s[h,1]
//   Cv[g] = sum_h W_pd[g,h]  * W_d[h,0]
//   K0[g] = W_ref@b_s + W_pd@b_d + b_ref + b_pd
// Also zero-initialize recurrent state (h, c, dec_in) every launch
// (determinism across graph replays).
// ---------------------------------------------------------------------------
__global__ void k_pre(const float* __restrict__ W_s, const float* __restrict__ W_d,
                      const float* __restrict__ W_ref, const float* __restrict__ b_ref,
                      const float* __restrict__ W_pd, const float* __restrict__ b_pd,
                      const float* __restrict__ b_s, const float* __restrict__ b_d,
                      const float* __restrict__ W_att,
                      float* __restrict__ coef4, float* __restrict__ K0,
                      float* __restrict__ h, float* __restrict__ c,
                      float* __restrict__ dec_in) {
  const int g = threadIdx.x;  // 0..127
  float a = 0.f, bv = 0.f, cv = 0.f, k0 = 0.f;
  for (int hh = 0; hh < HSZ; ++hh) {
    const float wr = W_ref[g * HSZ + hh];
    const float wp = W_pd[g * HSZ + hh];
    a  = fmaf(wr, W_s[hh * 2 + 0], a);
    bv = fmaf(wr, W_s[hh * 2 + 1], bv);
    cv = fmaf(wp, W_d[hh], cv);
    k0 = fmaf(wr, b_s[hh], fmaf(wp, b_d[hh], k0));
  }
  k0 += b_ref[g] + b_pd[g];
  coef4[g * 4 + 0] = a;
  coef4[g * 4 + 1] = bv;
  coef4[g * 4 + 2] = cv;
  coef4[g * 4 + 3] = W_att[g];
  K0[g] = k0;
  for (int i = g; i < BSZ * HSZ; i += HSZ) { h[i] = 0.f; c[i] = 0.f; }
  if (g < BSZ * 2) dec_in[g] = 0.f;
}

// ---------------------------------------------------------------------------
// k_lstm: one workgroup (32 waves).
//   stage 1: LDS A-image [16 x 256]: rows 0..7 = [dec_hidden | h_prev],
//            rows 8..15 = 0 (batch padding so EXEC stays all-1s at WMMA).
//   stage 2: gates(16x512) = A(16x256) x B(256x512) via V_WMMA_F32_16X16X4_F32,
//            one 16x16 N-tile per wave, K in chunks of 4.
//   stage 3: LSTM elementwise update; h2 written back into LDS rows 0..7.
//   stage 4: qk(16x128) = h2(16x128) @ W_q^T via WMMA on waves 0..7,
//            with b_q + K0 folded into the D-tile store.
// ---------------------------------------------------------------------------
__global__ void __launch_bounds__(1024)
k_lstm(const float* __restrict__ W_dec, const float* __restrict__ b_dec,
       const float* __restrict__ W_ih, const float* __restrict__ W_hh,
       const float* __restrict__ b_ih, const float* __restrict__ b_hh,
       const float* __restrict__ W_q, const float* __restrict__ b_q,
       const float* __restrict__ K0,
       const float* __restrict__ dec_in,
       float* __restrict__ h_ws, float* __restrict__ c_ws,
       float* __restrict__ qk_ws) {
  __shared__ float Ald[16 * 256];
  __shared__ float gates[8 * 512];
  const int t = threadIdx.x;

  // stage 1
  for (int idx = t; idx < 16 * 256; idx += 1024) {
    const int row = idx >> 8;
    const int k = idx & 255;
    float v = 0.0f;
    if (row < 8) {
      if (k < HSZ)
        v = fmaf(W_dec[k * 2 + 0], dec_in[row * 2 + 0],
                 fmaf(W_dec[k * 2 + 1], dec_in[row * 2 + 1], b_dec[k]));
      else
        v = h_ws[row * HSZ + (k - HSZ)];
    }
    Ald[idx] = v;
  }
  __syncthreads();

  // stage 2: WMMA. A layout (f32 16x4): lane l holds M=l&15; VGPR v holds
  // K = 4*kc + v + 2*(l>=16). B layout mirrored: lane l holds N=l&15, same K
  // mapping. D layout (ISA-given): lane l -> N=l&15, VGPR v -> M=v+8*(l>>4).
  const int w = t >> 5;
  const int l = t & 31;
  const int mm = l & 15;
  const int kh = (l >> 4) << 1;         // 0 or 2
  const int n = (w << 4) + mm;          // output column for this lane

  v8f acc = {0.f, 0.f, 0.f, 0.f, 0.f, 0.f, 0.f, 0.f};
  for (int kc = 0; kc < 64; ++kc) {
    const int kb = (kc << 2) + kh;      // even, so kb and kb+1 on same side of 128
    v2f av, bvv;
    av[0] = Ald[mm * 256 + kb];
    av[1] = Ald[mm * 256 + kb + 1];
    const float* Wp = (kb < HSZ) ? (W_ih + n * HSZ + kb)
                                 : (W_hh + n * HSZ + (kb - HSZ));
    bvv[0] = Wp[0];
    bvv[1] = Wp[1];
    acc = __builtin_amdgcn_wmma_f32_16x16x4_f32(false, av, false, bvv,
                                                (short)0, acc, false, false);
  }
  const float bias = b_ih[n] + b_hh[n];
  if (l < 16) {  // rows M=0..7 live in lanes 0..15 (VGPR v = M)
#pragma unroll
    for (int v = 0; v < 8; ++v) gates[v * 512 + n] = acc[v] + bias;
  }
  __syncthreads();

  // stage 3: elementwise LSTM (torch gate order: i, f, g, o)
  const int m = t >> 7;
  const int j = t & 127;
  {
    const float gi = gates[m * 512 + j];
    const float gf = gates[m * 512 + 128 + j];
    const float gg = gates[m * 512 + 256 + j];
    const float go = gates[m * 512 + 384 + j];
    const float c2 = sigm_(gf) * c_ws[t] + sigm_(gi) * tanh_(gg);
    const float h2 = sigm_(go) * tanh_(c2);
    c_ws[t] = c2;
    h_ws[t] = h2;
    Ald[m * 256 + 128 + j] = h2;  // safe: all stage-2 reads completed at barrier
  }
  __syncthreads();

  // stage 4: qk = h2 @ W_q.T + b_q + K0 via WMMA on waves 0..7 (N = 0..127).
  // A-frags read Ald[.][128..255]; rows 8..15 there are still zero from
  // stage 1. The w<8 branch is wave-uniform -> EXEC all-1s inside.
  if (w < 8) {
    v8f qacc = {0.f, 0.f, 0.f, 0.f, 0.f, 0.f, 0.f, 0.f};
    for (int kc = 0; kc < 32; ++kc) {
      const int kb = (kc << 2) + kh;
      v2f av, bvv;
      av[0] = Ald[mm * 256 + 128 + kb];
      av[1] = Ald[mm * 256 + 128 + kb + 1];
      bvv[0] = W_q[n * HSZ + kb];
      bvv[1] = W_q[n * HSZ + kb + 1];
      qacc = __builtin_amdgcn_wmma_f32_16x16x4_f32(false, av, false, bvv,
                                                   (short)0, qacc, false, false);
    }
    if (l < 16) {
      const float qb = b_q[n] + K0[n];
#pragma unroll
      for (int v = 0; v < 8; ++v) qk_ws[v * HSZ + n] = qacc[v] + qb;
    }
  }
}

// ---------------------------------------------------------------------------
// k_attn: grid (32, 8), 256 threads. Thread owns one city s of batch b.
//   attn[b,s] = sum_g W_att[g] * tanh(qk[b,g] + A[g]x + Bv[g]y + Cv[g]d)
// then block-level (max, argmax, sum exp(a - max)) partials.
// Only 12 bytes of per-city input per step -> L2-resident, tanh-bound.
// ---------------------------------------------------------------------------
__global__ void __launch_bounds__(256)
k_attn(const float* __restrict__ stat, const float* __restrict__ dyn,
       const float* __restrict__ coef4, const float* __restrict__ qk,
       float* __restrict__ pmax, float* __restrict__ pidx,
       float* __restrict__ psum) {
  __shared__ float4 cf[HSZ];
  __shared__ float qs[HSZ];
  __shared__ float rv[256];
  __shared__ int ri[256];

  const int b = blockIdx.y;
  const int tid = threadIdx.x;
  if (tid < HSZ) {
    cf[tid] = ((const float4*)coef4)[tid];
    qs[tid] = qk[b * HSZ + tid];
  }
  __syncthreads();

  const int s = blockIdx.x * 256 + tid;
  const float x = stat[(b * 2 + 0) * SSZ + s];
  const float y = stat[(b * 2 + 1) * SSZ + s];
  const float d = dyn[b * SSZ + s];

  float acc = 0.f;
#pragma unroll 4
  for (int g = 0; g < HSZ; ++g) {
    const float4 cv = cf[g];
    const float tt = fmaf(cv.x, x, fmaf(cv.y, y, fmaf(cv.z, d, qs[g])));
    acc = fmaf(cv.w, tanh_(tt), acc);
  }

  // block max/argmax (first-occurrence tie-break, matching jnp.argmax)
  rv[tid] = acc;
  ri[tid] = s;
  __syncthreads();
  for (int off = 128; off > 0; off >>= 1) {
    if (tid < off) {
      const float v2 = rv[tid + off];
      const int i2 = ri[tid + off];
      if (v2 > rv[tid] || (v2 == rv[tid] && i2 < ri[tid])) { rv[tid] = v2; ri[tid] = i2; }
    }
    __syncthreads();
  }
  const float bmax = rv[0];
  const int bidx = ri[0];
  __syncthreads();

  rv[tid] = expf(acc - bmax);
  __syncthreads();
  for (int off = 128; off > 0; off >>= 1) {
    if (tid < off) rv[tid] += rv[tid + off];
    __syncthreads();
  }
  if (tid == 0) {
    const int p = b * 32 + blockIdx.x;
    pmax[p] = bmax;
    pidx[p] = (float)bidx;  // <= 8191, exact in f32
    psum[p] = rv[0];
  }
}

// ---------------------------------------------------------------------------
// k_reduce: one block per batch, 32 threads. Combine 32 block partials:
//   gm = max, gi = argmax (first occurrence), total = sum_i psum_i*exp(max_i-gm)
//   logp = log(max softmax) = -log(total)   (the +10000 mask shift cancels)
// Writes outputs and gathers next dec_in = static[:, :, ptr].
// ---------------------------------------------------------------------------
__global__ void __launch_bounds__(32)
k_reduce(const float* __restrict__ pmax, const float* __restrict__ pidx,
         const float* __restrict__ psum, const float* __restrict__ stat,
         float* __restrict__ dec_in, float* __restrict__ out, int step) {
  __shared__ float mv[32];
  __shared__ int mi[32];
  __shared__ float sv[32];
  const int b = blockIdx.x;
  const int tid = threadIdx.x;
  const int p = b * 32 + tid;
  const float mx = pmax[p];
  mv[tid] = mx;
  mi[tid] = (int)pidx[p];
  __syncthreads();
  for (int off = 16; off > 0; off >>= 1) {
    if (tid < off) {
      const float v2 = mv[tid + off];
      const int i2 = mi[tid + off];
      if (v2 > mv[tid] || (v2 == mv[tid] && i2 < mi[tid])) { mv[tid] = v2; mi[tid] = i2; }
    }
    __syncthreads();
  }
  const float gm = mv[0];
  const int gi = mi[0];
  sv[tid] = psum[p] * expf(mx - gm);
  __syncthreads();
  for (int off = 16; off > 0; off >>= 1) {
    if (tid < off) sv[tid] += sv[tid + off];
    __syncthreads();
  }
  if (tid == 0) {
    out[b * NSTEP + step] = (float)gi;                       // tour_idx.T
    out[BSZ * NSTEP + b * NSTEP + step] = -logf(sv[0]);      // tour_logp.T
    dec_in[b * 2 + 0] = stat[(b * 2 + 0) * SSZ + gi];
    dec_in[b * 2 + 1] = stat[(b * 2 + 1) * SSZ + gi];
  }
}

// ---------------------------------------------------------------------------
extern "C" void kernel_launch(void* const* d_in, const int* in_sizes, int n_in,
                              void* d_out, int out_size, void* d_ws, size_t ws_size,
                              hipStream_t stream) {
  const float* stat  = (const float*)d_in[0];
  const float* dyn   = (const float*)d_in[1];
  const float* W_s   = (const float*)d_in[2];
  const float* b_s   = (const float*)d_in[3];
  const float* W_d   = (const float*)d_in[4];
  const float* b_d   = (const float*)d_in[5];
  const float* W_dec = (const float*)d_in[6];
  const float* b_dec = (const float*)d_in[7];
  const float* W_ih  = (const float*)d_in[8];
  const float* W_hh  = (const float*)d_in[9];
  const float* b_ih  = (const float*)d_in[10];
  const float* b_hh  = (const float*)d_in[11];
  const float* W_ref = (const float*)d_in[12];
  const float* b_ref = (const float*)d_in[13];
  const float* W_pd  = (const float*)d_in[14];
  const float* b_pd  = (const float*)d_in[15];
  const float* W_q   = (const float*)d_in[16];
  const float* b_q   = (const float*)d_in[17];
  const float* W_att = (const float*)d_in[18];
  (void)in_sizes; (void)n_in; (void)out_size; (void)ws_size;

  float* ws    = (float*)d_ws;
  float* coef4 = ws;           // 512  (A,Bv,Cv,W_att interleaved as float4)
  float* K0    = ws + 512;     // 128
  float* h     = ws + 640;     // 1024
  float* c     = ws + 1664;    // 1024
  float* dec   = ws + 2688;    // 16
  float* qk    = ws + 2704;    // 1024
  float* pmax  = ws + 3728;    // 256
  float* pidx  = ws + 3984;    // 256
  float* psum  = ws + 4240;    // 256  (total ~18 KB)
  float* out   = (float*)d_out;

  k_pre<<<1, 128, 0, stream>>>(W_s, W_d, W_ref, b_ref, W_pd, b_pd, b_s, b_d,
                               W_att, coef4, K0, h, c, dec);
  for (int t = 0; t < NSTEP; ++t) {
    k_lstm<<<1, 1024, 0, stream>>>(W_dec, b_dec, W_ih, W_hh, b_ih, b_hh,
                                   W_q, b_q, K0, dec, h, c, qk);
    k_attn<<<dim3(32, 8), 256, 0, stream>>>(stat, dyn, coef4, qk,
                                            pmax, pidx, psum);
    k_reduce<<<8, 32, 0, stream>>>(pmax, pidx, psum, stat, dec, out, t);
  }
}